// MultiScaleDeformableAttention_34479997453181
// MI455X (gfx1250) — compile-verified
//
#include <hip/hip_runtime.h>
#include <hip/hip_bf16.h>

typedef __attribute__((ext_vector_type(2))) float v2f;
typedef __attribute__((ext_vector_type(8))) float v8f;

#define NQ     13125
#define BATCH  2
#define DMODEL 256
#define HEADS  8
#define HD     32
#define LVLS   3
#define PTS    4

__device__ __constant__ int c_H[LVLS] = {100, 50, 25};
__device__ __constant__ int c_W[LVLS] = {100, 50, 25};

// ---------------------------------------------------------------------------
// fp32 WMMA GEMM:  C[M,N] = A[M,256] * W[256,N] + bias[N]
// Block = 4 waves; each wave owns a 16-row tile; all waves share one
// 32-column W stripe staged in LDS (column-major, 256+2 padded -> ds_load_b64
// B-fragments with at most 2-way bank conflicts). A-fragments are prefetched
// one k-step ahead so v_wmma does not drain loadcnt to zero each step.
//   A-frag (16x4 f32): lane l -> row l%16, K = k + (l/16)*2 + {0,1}
//   B-frag (4x16 f32): lane l -> col l%16, same K pairing
//   C/D   (16x16 f32): vgpr j, lane l -> row j + 8*(l>=16), col l%16
// ---------------------------------------------------------------------------
template <int NT>   // NT = 2 -> 32-wide column stripe per block
__global__ void gemm_bias_wmma_f32(const float* __restrict__ A,
                                   const float* __restrict__ W,
                                   const float* __restrict__ bias,
                                   float* __restrict__ C,
                                   int M, int N) {
  constexpr int STRIPE = 16 * NT;          // 32
  constexpr int KP     = DMODEL + 2;       // padded K stride (floats)
  __shared__ float lw[STRIPE * KP];        // ~33 KB

  const int lane  = threadIdx.x;           // 0..31
  const int tid   = threadIdx.y * 32 + lane;
  const int nthr  = blockDim.y * 32;       // 128
  const int n0    = blockIdx.y * STRIPE;

  // ---- cooperative stage of W stripe into LDS (coalesced 128B bursts) ----
  for (int i = tid; i < STRIPE * DMODEL; i += nthr) {
    const int col = i & (STRIPE - 1);      // fastest: coalesced global read
    const int k   = i / STRIPE;
    lw[col * KP + k] = W[(long)k * N + n0 + col];
  }
  __syncthreads();

  const int mtile = blockIdx.x * blockDim.y + threadIdx.y;
  const int m0    = mtile * 16;
  if (m0 < M) {
    const int half = lane >> 4;            // 0 | 1
    const int sub  = lane & 15;            // A row / B,C col within 16-tile

    int rowA = m0 + sub;
    if (rowA > M - 1) rowA = M - 1;        // clamp loads, guard stores
    const float* Arow = A + (long)rowA * DMODEL + half * 2;

    const float* lcol[NT];
#pragma unroll
    for (int t = 0; t < NT; ++t)
      lcol[t] = lw + (t * 16 + sub) * KP + half * 2;

    v8f acc[NT];
#pragma unroll
    for (int t = 0; t < NT; ++t)
      acc[t] = v8f{0.f, 0.f, 0.f, 0.f, 0.f, 0.f, 0.f, 0.f};

    v2f a = *(const v2f*)(Arow);           // prefetch k = 0
#pragma unroll 8
    for (int k = 0; k < DMODEL; k += 4) {
      v2f a_next = a;
      if (k + 4 < DMODEL) a_next = *(const v2f*)(Arow + k + 4);  // pipeline A
#pragma unroll
      for (int t = 0; t < NT; ++t) {
        const v2f b = *(const v2f*)(lcol[t] + k);                // ds_load_b64
        acc[t] = __builtin_amdgcn_wmma_f32_16x16x4_f32(
            false, a, false, b, (short)0, acc[t], false, false);
      }
      a = a_next;
    }

#pragma unroll
    for (int t = 0; t < NT; ++t) {
      const int col = n0 + t * 16 + sub;
      const float bv = bias[col];
#pragma unroll
      for (int j = 0; j < 8; ++j) {
        const int row = m0 + j + half * 8;
        if (row < M) C[(long)row * N + col] = acc[t][j] + bv;
      }
    }
  }
}

// ---------------------------------------------------------------------------
// Deformable sampling: one wave32 per (b, q, head); lane = channel (HD==32).
// 12-way softmax per head, then 48 bilinear taps; every tap is a coalesced
// 128B read of L2-resident projected values.
// ---------------------------------------------------------------------------
__global__ void msda_sample_kernel(const float* __restrict__ ref_pts,  // (B,Nq,L,2)
                                   const float* __restrict__ offv,    // (B*Nq,192)
                                   const float* __restrict__ attnv,   // (B*Nq,96)
                                   const float* __restrict__ vp0,
                                   const float* __restrict__ vp1,
                                   const float* __restrict__ vp2,
                                   float* __restrict__ sampled)       // (B*Nq,256)
{
  const int lane = threadIdx.x & 31;
  const int wid  = blockIdx.x * (blockDim.x >> 5) + (threadIdx.x >> 5);
  const int total = BATCH * NQ * HEADS;
  if (wid >= total) return;

  const int h = wid % HEADS;
  const int q = (wid / HEADS) % NQ;
  const int b = wid / (HEADS * NQ);
  const long row = (long)b * NQ + q;

  float lg[LVLS * PTS];
  const float* ap = attnv + row * (HEADS * LVLS * PTS) + h * (LVLS * PTS);
  float mx = -3.0e38f;
#pragma unroll
  for (int i = 0; i < LVLS * PTS; ++i) { lg[i] = ap[i]; mx = fmaxf(mx, lg[i]); }
  float sum = 0.f;
#pragma unroll
  for (int i = 0; i < LVLS * PTS; ++i) { lg[i] = __expf(lg[i] - mx); sum += lg[i]; }
  const float inv = 1.f / sum;

  const float* offp = offv + row * (HEADS * LVLS * PTS * 2) + h * (LVLS * PTS * 2);
  const float* vps[LVLS] = {vp0, vp1, vp2};

  float acc = 0.f;
#pragma unroll
  for (int lvl = 0; lvl < LVLS; ++lvl) {
    const int Wl = c_W[lvl];
    const int Hl = c_H[lvl];
    const float refx = ref_pts[(row * LVLS + lvl) * 2 + 0];
    const float refy = ref_pts[(row * LVLS + lvl) * 2 + 1];
    const float* vp = vps[lvl] + ((long)b * Hl * Wl) * DMODEL + h * HD + lane;
#pragma unroll
    for (int p = 0; p < PTS; ++p) {
      const float ox = offp[(lvl * PTS + p) * 2 + 0];
      const float oy = offp[(lvl * PTS + p) * 2 + 1];
      const float x = refx * (float)Wl + ox - 0.5f;   // align_corners=False
      const float y = refy * (float)Hl + oy - 0.5f;
      const float x0f = floorf(x), y0f = floorf(y);
      const int x0 = (int)x0f, y0 = (int)y0f;
      const float wx1 = x - x0f, wy1 = y - y0f;
      const float aw = lg[lvl * PTS + p] * inv;
#pragma unroll
      for (int dy = 0; dy < 2; ++dy) {
#pragma unroll
        for (int dx = 0; dx < 2; ++dx) {
          const int xi = x0 + dx, yi = y0 + dy;
          if (xi >= 0 && xi < Wl && yi >= 0 && yi < Hl) {
            const float w = (dx ? wx1 : 1.f - wx1) * (dy ? wy1 : 1.f - wy1);
            acc += aw * w * vp[(long)(yi * Wl + xi) * DMODEL];
          }
        }
      }
    }
  }
  sampled[row * DMODEL + h * HD + lane] = acc;
}

// ---------------------------------------------------------------------------
extern "C" void kernel_launch(void* const* d_in, const int* in_sizes, int n_in,
                              void* d_out, int out_size, void* d_ws, size_t ws_size,
                              hipStream_t stream) {
  const float* query   = (const float*)d_in[0];
  const float* refpts  = (const float*)d_in[1];
  const float* value0  = (const float*)d_in[2];
  const float* value1  = (const float*)d_in[3];
  const float* value2  = (const float*)d_in[4];
  const float* W_val   = (const float*)d_in[5];
  const float* b_val   = (const float*)d_in[6];
  const float* W_off   = (const float*)d_in[7];
  const float* b_off   = (const float*)d_in[8];
  const float* W_attn  = (const float*)d_in[9];
  const float* b_attn  = (const float*)d_in[10];
  const float* W_out   = (const float*)d_in[11];
  const float* b_out   = (const float*)d_in[12];
  float* out = (float*)d_out;

  const int Mq  = BATCH * NQ;       // 26250
  const int Mv0 = BATCH * 10000;    // 20000
  const int Mv1 = BATCH * 2500;     //  5000
  const int Mv2 = BATCH * 625;      //  1250

  float* ws      = (float*)d_ws;
  float* ws_off  = ws;                                   // Mq * 192
  float* ws_attn = ws_off  + (size_t)Mq * 192;           // Mq * 96
  float* vp0     = ws_attn + (size_t)Mq * 96;            // Mv0 * 256
  float* vp1     = vp0     + (size_t)Mv0 * DMODEL;       // Mv1 * 256
  float* vp2     = vp1     + (size_t)Mv1 * DMODEL;       // Mv2 * 256
  float* smp     = vp2     + (size_t)Mv2 * DMODEL;       // Mq * 256
  const size_t need = ((size_t)Mq * 192 + (size_t)Mq * 96 +
                       (size_t)(Mv0 + Mv1 + Mv2 + Mq) * DMODEL) * sizeof(float);
  if (ws_size < need) return;

  const dim3 blk(32, 4);  // 4 waves / block, shared 32-col W stripe in LDS

  // value projection per level (N = 256 -> 8 column stripes)
  gemm_bias_wmma_f32<2><<<dim3((Mv0 + 63) / 64, DMODEL / 32), blk, 0, stream>>>(
      value0, W_val, b_val, vp0, Mv0, DMODEL);
  gemm_bias_wmma_f32<2><<<dim3((Mv1 + 63) / 64, DMODEL / 32), blk, 0, stream>>>(
      value1, W_val, b_val, vp1, Mv1, DMODEL);
  gemm_bias_wmma_f32<2><<<dim3((Mv2 + 63) / 64, DMODEL / 32), blk, 0, stream>>>(
      value2, W_val, b_val, vp2, Mv2, DMODEL);

  // sampling offsets (N = 192 -> 6 stripes)
  gemm_bias_wmma_f32<2><<<dim3((Mq + 63) / 64, 192 / 32), blk, 0, stream>>>(
      query, W_off, b_off, ws_off, Mq, 192);

  // attention logits (N = 96 -> 3 stripes)
  gemm_bias_wmma_f32<2><<<dim3((Mq + 63) / 64, 96 / 32), blk, 0, stream>>>(
      query, W_attn, b_attn, ws_attn, Mq, 96);

  // deformable bilinear sampling: one wave per (b, q, head)
  {
    const int totalWaves = BATCH * NQ * HEADS;          // 210000
    const int wavesPerBlk = 8;                          // 256 threads
    const int grid = (totalWaves + wavesPerBlk - 1) / wavesPerBlk;
    msda_sample_kernel<<<grid, wavesPerBlk * 32, 0, stream>>>(
        refpts, ws_off, ws_attn, vp0, vp1, vp2, smp);
  }

  // output projection (N = 256)
  gemm_bias_wmma_f32<2><<<dim3((Mq + 63) / 64, DMODEL / 32), blk, 0, stream>>>(
      smp, W_out, b_out, out, Mq, DMODEL);
}